// EnhancedMultiGPULoss_88210038325435
// MI455X (gfx1250) — compile-verified
//
#include <hip/hip_runtime.h>
#include <math.h>

typedef float v2f __attribute__((ext_vector_type(2)));
typedef float v8f __attribute__((ext_vector_type(8)));

// ws float slots
#define WS_CHAM0    0
#define WS_CHAM1    1
#define WS_COVSUM   2   // 8 slots
#define WS_COVCNT   10  // 8 slots
#define WS_REP      18
#define WS_SUMD     19  // 8 slots
#define WS_SUMD2    27  // 8 slots
#define WS_CNT      35  // 8 slots
#define WS_SMOOTH   43
#define WS_MEANSTD  44  // 8 slots

__global__ void init_ws(float* ws) {
    if (threadIdx.x < 64) ws[threadIdx.x] = 0.0f;
}

// One wave per 16-row tile of A. Sweep all 16-col tiles of B.
// d2[m][n] = |a_m|^2 + |b_n|^2 - 2 (a_m . b_n).
// A operand is pre-scaled by -2, so WMMA accumulates -2*dot directly; the
// per-row |a_m|^2 and the zero-clamp are hoisted after the column sweep:
//   min_n max(na + nb - 2dot, 0) = max(na + min_n(nb - 2dot), 0).
// mode 0: atomicAdd sum of row-min d2 into ws[slot]   (chamfer direction)
// mode 1: coverage: masked sum of sqrt(rowmin d2) per batch + count per batch
__global__ __launch_bounds__(256) void wmma_rowmin(
    const float* __restrict__ A, const float* __restrict__ Bm,
    int NB, int N, int M, int mode, int slot, float* __restrict__ ws)
{
    const int lane       = threadIdx.x & 31;
    const int wave       = (blockIdx.x * 256 + (int)threadIdx.x) >> 5;
    const int tilesPerB  = N >> 4;
    if (wave >= NB * tilesPerB) return;           // uniform across wave
    const int b  = wave / tilesPerB;
    const int r0 = (wave % tilesPerB) << 4;
    const int hi = lane >> 4;                      // 0: K=0,1 half; 1: K=2,3 half
    const int l  = lane & 15;

    const float* __restrict__ Ab = A  + (size_t)b * N * 3;
    const float* __restrict__ Bb = Bm + (size_t)b * M * 3;

    // ---- A operand: 16x4 f32 tile, K = -2*(x,y,z,0) ----
    // lanes 0-15: (-2x,-2y) of row l ; lanes 16-31: (-2z,0) of row l
    float ax = Ab[3 * (r0 + l) + 0];
    float ay = Ab[3 * (r0 + l) + 1];
    float az = Ab[3 * (r0 + l) + 2];
    v2f aop;
    aop[0] = -2.0f * (hi ? az : ax);
    aop[1] = hi ? 0.0f : (-2.0f * ay);

    // per-lane row norms / validity for rows r0 + v + 8*hi (matches D-tile layout);
    // na[v] is identical across the 16 lanes of each half.
    float na[8], valid[8];
#pragma unroll
    for (int v = 0; v < 8; ++v) {
        int r = r0 + v + 8 * hi;
        float x = Ab[3 * r + 0], y = Ab[3 * r + 1], z = Ab[3 * r + 2];
        na[v]    = x * x + y * y + z * z;
        valid[v] = (fabsf(x) + fabsf(y) + fabsf(z) > 1e-6f) ? 1.0f : 0.0f;
    }

    // running min of (nb - 2*dot) per row
    float rmin[8];
#pragma unroll
    for (int v = 0; v < 8; ++v) rmin[v] = 3.0e38f;

    const int ctiles = M >> 4;
    for (int ct = 0; ct < ctiles; ++ct) {
        const int c0 = ct << 4;
        // ---- B operand: 4x16 f32 tile (mirrored layout of A, unscaled) ----
        float bx = Bb[3 * (c0 + l) + 0];
        float by = Bb[3 * (c0 + l) + 1];
        float bz = Bb[3 * (c0 + l) + 2];
        float nb = bx * bx + by * by + bz * bz;
        v2f bop;
        bop[0] = hi ? bz : bx;
        bop[1] = hi ? 0.0f : by;

        v8f acc = {0.f, 0.f, 0.f, 0.f, 0.f, 0.f, 0.f, 0.f};
        // (neg_a, A, neg_b, B, c_mod, C, reuse_a, reuse_b) ; acc = -2*dot
        acc = __builtin_amdgcn_wmma_f32_16x16x4_f32(
            false, aop, false, bop, (short)0, acc, false, false);

#pragma unroll
        for (int v = 0; v < 8; ++v)
            rmin[v] = fminf(rmin[v], nb + acc[v]);
    }

    // cross-lane min across each 16-lane half (butterfly)
#pragma unroll
    for (int v = 0; v < 8; ++v) {
#pragma unroll
        for (int m = 1; m <= 8; m <<= 1)
            rmin[v] = fminf(rmin[v], __shfl_xor(rmin[v], m, 32));
    }

    // deferred |a|^2 add + zero clamp
#pragma unroll
    for (int v = 0; v < 8; ++v)
        rmin[v] = fmaxf(na[v] + rmin[v], 0.0f);

    if (l == 0) {  // lanes 0 (rows 0-7) and 16 (rows 8-15)
        if (mode == 0) {
            float s = 0.0f;
#pragma unroll
            for (int v = 0; v < 8; ++v) s += rmin[v];
            atomicAdd(&ws[slot], s);
        } else {
            float s = 0.0f, c = 0.0f;
#pragma unroll
            for (int v = 0; v < 8; ++v) {
                c += valid[v];
                s += valid[v] * sqrtf(rmin[v]);
            }
            atomicAdd(&ws[WS_COVSUM + b], s);
            atomicAdd(&ws[WS_COVCNT + b], c);
        }
    }
}

__device__ __forceinline__ float block_reduce_sum(float v, float* red) {
    red[threadIdx.x] = v;
    __syncthreads();
    for (int s = 128; s > 0; s >>= 1) {
        if ((int)threadIdx.x < s) red[threadIdx.x] += red[threadIdx.x + s];
        __syncthreads();
    }
    float r = red[0];
    __syncthreads();
    return r;
}

// Repulsion 8-NN distances + diversity distance statistics, thread-per-point.
__global__ __launch_bounds__(256) void selfdist_kernel(
    const float* __restrict__ P, int NB, int N, float* __restrict__ ws)
{
    __shared__ float red[256];
    const int t = blockIdx.x * 256 + threadIdx.x;
    const int b = t / N;
    const int i = t % N;
    const float* __restrict__ Pb = P + (size_t)b * N * 3;
    const float px = Pb[3 * i + 0], py = Pb[3 * i + 1], pz = Pb[3 * i + 2];

    float knn[8];
#pragma unroll
    for (int k = 0; k < 8; ++k) knn[k] = 3.0e38f;

    float s1 = 0.0f, s2 = 0.0f, c = 0.0f;
    for (int j = 0; j < N; ++j) {
        float dx = px - Pb[3 * j + 0];
        float dy = py - Pb[3 * j + 1];
        float dz = pz - Pb[3 * j + 2];
        float d2 = dx * dx + dy * dy + dz * dz;
        float d  = sqrtf(d2);
        s1 += d;
        s2 += d2;
        if (d2 > 0.0f) c += 1.0f;
        if (j != i && d < knn[7]) {          // diagonal excluded (eye*1e6)
            int p = 7;
#pragma unroll
            for (int q = 7; q > 0; --q) {
                if (knn[q - 1] > d) { knn[q] = knn[q - 1]; p = q - 1; }
            }
            knn[p] = d;
        }
    }

    float rep = 0.0f;
#pragma unroll
    for (int k = 0; k < 8; ++k) {
        float dk = knn[k];
        rep += 10.0f * fmaxf(0.005f - dk, 0.0f)
             +  5.0f * fmaxf(0.01f  - dk, 0.0f)
             +         fmaxf(0.02f  - dk, 0.0f);
    }

    // block is entirely inside one batch (256 | N)
    float rep_r = block_reduce_sum(rep, red);
    float s1_r  = block_reduce_sum(s1, red);
    float s2_r  = block_reduce_sum(s2, red);
    float c_r   = block_reduce_sum(c, red);
    if (threadIdx.x == 0) {
        atomicAdd(&ws[WS_REP], rep_r);
        atomicAdd(&ws[WS_SUMD  + b], s1_r);
        atomicAdd(&ws[WS_SUMD2 + b], s2_r);
        atomicAdd(&ws[WS_CNT   + b], c_r);
    }
}

// Smoothness: 16-NN (incl. self) for first n points per batch; local variance.
__global__ void smooth_kernel(const float* __restrict__ P,
                              int NB, int N, int n, float* __restrict__ ws)
{
    const int t = blockIdx.x * blockDim.x + threadIdx.x;
    if (t >= NB * n) return;
    const int b = t / n;
    const int i = t % n;
    const float* __restrict__ Pb = P + (size_t)b * N * 3;
    const float px = Pb[3 * i + 0], py = Pb[3 * i + 1], pz = Pb[3 * i + 2];

    float kd[16];
    int   ki[16];
#pragma unroll
    for (int k = 0; k < 16; ++k) { kd[k] = 3.0e38f; ki[k] = 0; }

    for (int j = 0; j < N; ++j) {
        float dx = px - Pb[3 * j + 0];
        float dy = py - Pb[3 * j + 1];
        float dz = pz - Pb[3 * j + 2];
        float d2 = dx * dx + dy * dy + dz * dz;
        if (d2 < kd[15]) {
            int p = 15;
#pragma unroll
            for (int q = 15; q > 0; --q) {
                if (kd[q - 1] > d2) { kd[q] = kd[q - 1]; ki[q] = ki[q - 1]; p = q - 1; }
            }
            kd[p] = d2;
            ki[p] = j;
        }
    }

    float sx = 0, sy = 0, sz = 0, sxx = 0, syy = 0, szz = 0;
#pragma unroll
    for (int k = 0; k < 16; ++k) {
        int j = ki[k];
        float x = Pb[3 * j + 0], y = Pb[3 * j + 1], z = Pb[3 * j + 2];
        sx += x; sy += y; sz += z;
        sxx += x * x; syy += y * y; szz += z * z;
    }
    const float inv16 = 1.0f / 16.0f;
    float dev = (sxx - sx * sx * inv16)
              + (syy - sy * sy * inv16)
              + (szz - sz * sz * inv16);
    float var = fmaxf(dev, 0.0f) / 47.0f;   // SMOOTH_K*3 - 1
    atomicAdd(&ws[WS_SMOOTH], var);
}

// Per-batch coordinate std (ddof=1), mean over dims -> ws[WS_MEANSTD + b]
__global__ __launch_bounds__(256) void std_kernel(
    const float* __restrict__ P, int N, float* __restrict__ ws)
{
    __shared__ float red[256];
    const int b = blockIdx.x;
    const float* __restrict__ Pb = P + (size_t)b * N * 3;
    float sx = 0, sy = 0, sz = 0, sxx = 0, syy = 0, szz = 0;
    for (int i = threadIdx.x; i < N; i += 256) {
        float x = Pb[3 * i + 0], y = Pb[3 * i + 1], z = Pb[3 * i + 2];
        sx += x; sy += y; sz += z;
        sxx += x * x; syy += y * y; szz += z * z;
    }
    sx  = block_reduce_sum(sx, red);
    sy  = block_reduce_sum(sy, red);
    sz  = block_reduce_sum(sz, red);
    sxx = block_reduce_sum(sxx, red);
    syy = block_reduce_sum(syy, red);
    szz = block_reduce_sum(szz, red);
    if (threadIdx.x == 0) {
        float fN = (float)N;
        float vx = fmaxf(sxx - sx * sx / fN, 0.0f) / (fN - 1.0f);
        float vy = fmaxf(syy - sy * sy / fN, 0.0f) / (fN - 1.0f);
        float vz = fmaxf(szz - sz * sz / fN, 0.0f) / (fN - 1.0f);
        ws[WS_MEANSTD + b] = (sqrtf(vx) + sqrtf(vy) + sqrtf(vz)) * (1.0f / 3.0f);
    }
}

__global__ void finalize_kernel(const float* __restrict__ ws, float* __restrict__ out,
                                int B, int N, int M, int P, int n)
{
    if (threadIdx.x != 0 || blockIdx.x != 0) return;
    float total = 0.0f;
    // chamfer (w = 1.0)
    total += ws[WS_CHAM0] / (float)(B * N) + ws[WS_CHAM1] / (float)(B * M);
    // coverage (w = 0.2)
    float cov = 0.0f;
    for (int b = 0; b < B; ++b) {
        float c  = ws[WS_COVCNT + b];
        float pb = ws[WS_COVSUM + b] / fmaxf(c, 1.0f);
        cov += (c > 0.0f) ? pb : 0.0f;
    }
    total += 0.2f * (cov / (float)B);
    // repulsion (w = 0.2), mean over (B,N,8)
    total += 0.2f * (ws[WS_REP] / (float)(B * N * 8));
    // smoothness (w = 0.05), sum/(B*n)
    total += 0.05f * (ws[WS_SMOOTH] / (float)(B * n));
    // diversity (w = 0.3): (mean_b relu(0.3-mean_std) + sum_b relu(0.1-dist_std)) / B
    float p1 = 0.0f, p2 = 0.0f;
    for (int b = 0; b < B; ++b) {
        p1 += fmaxf(0.3f - ws[WS_MEANSTD + b], 0.0f);
        float cnt  = ws[WS_CNT + b];
        float sd   = ws[WS_SUMD + b];
        float sd2  = ws[WS_SUMD2 + b];
        float mean = sd / fmaxf(cnt, 1.0f);
        float var  = (sd2 - 2.0f * mean * sd + cnt * mean * mean) / fmaxf(cnt - 1.0f, 1.0f);
        p2 += fmaxf(0.1f - sqrtf(fmaxf(var, 0.0f)), 0.0f);
    }
    p1 /= (float)B;
    total += 0.3f * ((p1 + p2) / (float)B);
    out[0] = total;
}

extern "C" void kernel_launch(void* const* d_in, const int* in_sizes, int n_in,
                              void* d_out, int out_size, void* d_ws, size_t ws_size,
                              hipStream_t stream)
{
    const float* pred    = (const float*)d_in[0];  // (8,4096,3)
    const float* gt      = (const float*)d_in[1];  // (8,4096,3)
    const float* partial = (const float*)d_in[2];  // (8,2048,3)
    float* out = (float*)d_out;
    float* ws  = (float*)d_ws;

    const int B = 8;
    const int N = in_sizes[0] / (B * 3);   // 4096
    const int M = in_sizes[1] / (B * 3);   // 4096
    const int P = in_sizes[2] / (B * 3);   // 2048
    const int n = (N < 500) ? N : 500;

    init_ws<<<1, 64, 0, stream>>>(ws);

    // chamfer: pred->gt row-min d^2, and gt->pred
    {
        int tiles  = B * (N >> 4);
        int blocks = (tiles + 7) / 8;
        wmma_rowmin<<<blocks, 256, 0, stream>>>(pred, gt, B, N, M, 0, WS_CHAM0, ws);
    }
    {
        int tiles  = B * (M >> 4);
        int blocks = (tiles + 7) / 8;
        wmma_rowmin<<<blocks, 256, 0, stream>>>(gt, pred, B, M, N, 0, WS_CHAM1, ws);
    }
    // coverage: partial->pred masked row-min distance
    {
        int tiles  = B * (P >> 4);
        int blocks = (tiles + 7) / 8;
        wmma_rowmin<<<blocks, 256, 0, stream>>>(partial, pred, B, P, N, 1, 0, ws);
    }
    // repulsion + diversity distance stats
    selfdist_kernel<<<(B * N) / 256, 256, 0, stream>>>(pred, B, N, ws);
    // smoothness
    smooth_kernel<<<(B * n + 255) / 256, 256, 0, stream>>>(pred, B, N, n, ws);
    // per-batch coordinate std
    std_kernel<<<B, 256, 0, stream>>>(pred, N, ws);
    // combine
    finalize_kernel<<<1, 1, 0, stream>>>(ws, out, B, N, M, P, n);
}